// Line_29205777613284
// MI455X (gfx1250) — compile-verified
//
#include <hip/hip_runtime.h>

typedef __attribute__((ext_vector_type(2))) float v2f;
typedef __attribute__((ext_vector_type(8))) float v8f;

#define EMB  128
#define KNEG 5

__device__ __forceinline__ float log_sigmoid(float x) {
    // log(sigmoid(x)) = min(x,0) - log1p(exp(-|x|))   (numerically stable)
    return fminf(x, 0.0f) - log1pf(__expf(-fabsf(x)));
}

// Extract element r (0..7) from a v8f accumulator with a cndmask chain.
__device__ __forceinline__ float diag_sel(v8f v, int r) {
    float x = v[0];
    x = (r == 1) ? v[1] : x;
    x = (r == 2) ? v[2] : x;
    x = (r == 3) ? v[3] : x;
    x = (r == 4) ? v[4] : x;
    x = (r == 5) ? v[5] : x;
    x = (r == 6) ? v[6] : x;
    x = (r == 7) ? v[7] : x;
    return x;
}

__global__ void sgns_partial_kernel(const float* __restrict__ second,
                                    const float* __restrict__ context,
                                    const int*   __restrict__ vi,
                                    const int*   __restrict__ vj,
                                    const int*   __restrict__ negs,
                                    float*       __restrict__ partial,
                                    int batch) {
    const int lane        = threadIdx.x & 31;
    const int wid         = threadIdx.x >> 5;
    const int wavesPerBlk = blockDim.x >> 5;
    const int gw          = blockIdx.x * wavesPerBlk + wid;
    const int nwaves      = gridDim.x * wavesPerBlk;
    const int groups      = batch >> 4;     // 16 batch elements per wave-group

    // WMMA f32 16x16x4 fragment addressing:
    //   A (16x4, M x K): lane holds row M = lane&15, K-pair base (lane>>4)*2
    //   B (4x16, K x N): lane holds col N = lane&15, K-pair base (lane>>4)*2
    // -> identical float2 gather pattern for both operands.
    const int n    = lane & 15;
    const int koff = (lane >> 4) << 1;      // 0 or 2

    float local = 0.0f;

    for (int g = gw; g < groups; g += nwaves) {
        const int base = g << 4;

        const float* pA = second + (size_t)vi[base + n] * EMB + koff;
        const float* pB[KNEG + 1];
        pB[0] = context + (size_t)vj[base + n] * EMB + koff;
#pragma unroll
        for (int k = 0; k < KNEG; ++k)
            pB[k + 1] = context + (size_t)negs[k * batch + base + n] * EMB + koff;

        v8f acc[KNEG + 1];
#pragma unroll
        for (int t = 0; t < KNEG + 1; ++t) {
            v8f z = {};
            acc[t] = z;
        }

#pragma unroll 4
        for (int kc = 0; kc < EMB / 4; ++kc) {
            v2f a = *(const v2f*)(pA + kc * 4);
#pragma unroll
            for (int t = 0; t < KNEG + 1; ++t) {
                v2f b = *(const v2f*)(pB[t] + kc * 4);
                // D = A x B + C  (f32 16x16x4), 8 args:
                // (neg_a, A, neg_b, B, c_mod, C, reuse_a, reuse_b)
                acc[t] = __builtin_amdgcn_wmma_f32_16x16x4_f32(
                    false, a, false, b, (short)0, acc[t], false, false);
            }
        }

        // Diagonal of 16x16 f32 D tile:
        //   lanes 0..7  hold diag m = lane      in acc VGPR r = lane
        //   lanes 24..31 hold diag m = lane-16  in acc VGPR r = lane&7
        const int  r     = lane & 7;
        const bool valid = (lane < 8) || (lane >= 24);
        float c = 0.0f;
        if (valid) {
            c = log_sigmoid(diag_sel(acc[0], r));           // positive score
#pragma unroll
            for (int t = 1; t <= KNEG; ++t)
                c += log_sigmoid(-diag_sel(acc[t], r));     // negative scores
        }
        local += c;
    }

    // wave32 reduction
#pragma unroll
    for (int off = 16; off > 0; off >>= 1)
        local += __shfl_xor(local, off, 32);

    __shared__ float ssum[32];
    if (lane == 0) ssum[wid] = local;
    __syncthreads();
    if (threadIdx.x == 0) {
        float s = 0.0f;
        for (int w = 0; w < wavesPerBlk; ++w) s += ssum[w];
        partial[blockIdx.x] = s;
    }
}

__global__ void sgns_finalize_kernel(const float* __restrict__ partial, int np,
                                     float* __restrict__ out, float invB) {
    __shared__ float red[256];
    float s = 0.0f;
    for (int i = threadIdx.x; i < np; i += blockDim.x) s += partial[i];
    red[threadIdx.x] = s;
    __syncthreads();
    for (int h = 128; h > 0; h >>= 1) {
        if ((int)threadIdx.x < h) red[threadIdx.x] += red[threadIdx.x + h];
        __syncthreads();
    }
    if (threadIdx.x == 0) out[0] = -red[0] * invB;
}

extern "C" void kernel_launch(void* const* d_in, const int* in_sizes, int n_in,
                              void* d_out, int out_size, void* d_ws, size_t ws_size,
                              hipStream_t stream) {
    const float* second  = (const float*)d_in[0];   // [DICT, 128] f32
    const float* context = (const float*)d_in[1];   // [DICT, 128] f32
    const int*   vi      = (const int*)d_in[2];     // [B]
    const int*   vj      = (const int*)d_in[3];     // [B]
    const int*   negs    = (const int*)d_in[4];     // [5, B]
    float*       out     = (float*)d_out;           // scalar f32
    float*       partial = (float*)d_ws;            // one f32 per block

    const int batch  = in_sizes[2];
    const int groups = batch >> 4;                  // 16 batch elems / wave
    const int wavesPerBlk = 8;                      // 256 threads, wave32
    int nblk = (groups + wavesPerBlk - 1) / wavesPerBlk;
    if (nblk > 1024) nblk = 1024;
    if (nblk < 1)    nblk = 1;

    sgns_partial_kernel<<<nblk, wavesPerBlk * 32, 0, stream>>>(
        second, context, vi, vj, negs, partial, batch);
    sgns_finalize_kernel<<<1, 256, 0, stream>>>(
        partial, nblk, out, 1.0f / (float)batch);
}